// Conv_RFF_24438363914658
// MI455X (gfx1250) — compile-verified
//
#include <hip/hip_runtime.h>
#include <hip/hip_bf16.h>

// ---------------------------------------------------------------------------
// Conv + Random-Fourier-Feature kernel for MI455X (gfx1250, wave32, WMMA)
//
//   phi_half[b, mc*32+o, y, x] = sum_{k=0..143} W[k][mc*32+o] * X[b,k,y,x]
//     where k = (kh*3+kw)*16 + ic,  X gathers x[b, ic, 2y-1+kh, 2x-1+kw]
//     W[k][mc*32+o] = eps[mc,k,o]*exp(0.5*logsig[k,o]) + mean[k,o]
//   out = concat(scale*cos(phi_half), scale*sin(phi_half)) per (mc) group
//
// GEMM mapping (V_WMMA_F32_16X16X4_F32, D = A x B + C):
//   A (16x4)  M = channel, K = k      -> pre-packed per-lane in scratch
//   B (4x16)  K = k, N = spatial pos  -> gathered b64 from padded-NHWC x
//   C (16x16) M = channel, N = pos    -> coalesced stores in x
// ---------------------------------------------------------------------------

typedef float v2f __attribute__((ext_vector_type(2)));
typedef float v8f __attribute__((ext_vector_type(8)));

#define XPAD_ELEMS (8 * 130 * 130 * 16)   // padded NHWC input, fp32
#define APACK_F2   (20 * 36 * 32)         // 20 ch-tiles * 36 ksteps * 32 lanes (float2)

// ---------------------------------------------------------------------------
// Kernel 0: NCHW -> padded NHWC.  One thread per (b, iy, ix): gathers 16
// channels (coalesced across lanes in ix), writes 64B contiguous.
// ---------------------------------------------------------------------------
__global__ __launch_bounds__(256) void rff_pad_nhwc(const float* __restrict__ x,
                                                    float* __restrict__ xpad) {
    int t = blockIdx.x * 256 + threadIdx.x;
    if (t >= 8 * 130 * 130) return;
    int ix = t % 130;
    int r  = t / 130;
    int iy = r % 130;
    int b  = r / 130;
    bool inb = (iy >= 1) && (iy <= 128) && (ix >= 1) && (ix <= 128);
    const float* xs = x + (((size_t)b * 16) * 128 + (iy - 1)) * 128 + (ix - 1);
    float v[16];
#pragma unroll
    for (int c = 0; c < 16; ++c) v[c] = inb ? xs[(size_t)c * 16384] : 0.0f;
    float4* dst = (float4*)(xpad + (size_t)t * 16);
#pragma unroll
    for (int q = 0; q < 4; ++q)
        dst[q] = make_float4(v[4 * q], v[4 * q + 1], v[4 * q + 2], v[4 * q + 3]);
}

// ---------------------------------------------------------------------------
// Kernel 1: reparameterize weights and pack A-fragments.
// Layout: apack[(mt*36 + ks)*32 + lane] = float2{ W[k][ch], W[k+1][ch] }
//   ch = mt*16 + (lane&15),  k = ks*4 + (lane>>4)*2
// Matches the 32-bit 16x4 A-matrix VGPR layout (lanes 0-15: K=0,1; 16-31: K=2,3).
// ---------------------------------------------------------------------------
__global__ __launch_bounds__(256) void rff_pack_w(const float* __restrict__ mean,
                                                  const float* __restrict__ logs,
                                                  const float* __restrict__ eps,
                                                  float* __restrict__ apack) {
    int t = blockIdx.x * 256 + threadIdx.x;
    if (t >= APACK_F2) return;
    int lane = t & 31;
    int r    = t >> 5;
    int ks   = r % 36;
    int mt   = r / 36;                       // 0..19
    int k    = ks * 4 + ((lane >> 4) << 1);  // 0..142, even
    int m    = lane & 15;
    int ch   = mt * 16 + m;                  // 0..319
    int mc   = ch >> 5;
    int o    = ch & 31;
    int i0   = k * 32 + o;                   // [144][32] index
    int ie   = (mc * 144 + k) * 32 + o;      // [10][144][32] index
    float2 w;
    w.x = eps[ie]      * __expf(0.5f * logs[i0])      + mean[i0];
    w.y = eps[ie + 32] * __expf(0.5f * logs[i0 + 32]) + mean[i0 + 32];
    ((float2*)apack)[t] = w;
}

// ---------------------------------------------------------------------------
// Kernel 2: implicit-GEMM conv with fp32 WMMA + cos/sin epilogue.
// Each wave: 32 channels (2 M-tiles) x 32 positions (2 N-tiles).
// 10 ch-groups * 1024 pos-groups = 10240 waves = 1280 blocks * 8 waves.
// ---------------------------------------------------------------------------
__global__ __launch_bounds__(256) void rff_conv_main(const float* __restrict__ xpad,
                                                     const float* __restrict__ apack,
                                                     const float* __restrict__ theta,
                                                     float* __restrict__ out) {
    const int lane = threadIdx.x & 31;
    const int wid  = threadIdx.x >> 5;
    const int task = blockIdx.x * 8 + wid;     // 0..10239
    const int cg   = task >> 10;               // 0..9   channel group (32 ch)
    const int pg   = task & 1023;              // 0..1023 position group (32 pos)
    const int b    = pg >> 7;                  // 0..7
    const int pb   = (pg & 127) << 5;          // 0..4064, multiple of 32

    const int n  = lane & 15;                  // N index (position in tile)
    const int kp = lane >> 4;                  // K half select
    const int p0 = pb + n;
    const int p1 = p0 + 16;
    const int y0 = p0 >> 6, x0 = p0 & 63;
    const int y1 = p1 >> 6, x1 = p1 & 63;

    const float* xb  = xpad + (size_t)b * (130 * 130 * 16);
    const float* ap0 = apack + (size_t)((cg * 2)     * 36) * 64 + lane * 2;
    const float* ap1 = apack + (size_t)((cg * 2 + 1) * 36) * 64 + lane * 2;

    v8f acc00 = {}; v8f acc01 = {}; v8f acc10 = {}; v8f acc11 = {};

#pragma unroll 4
    for (int ks = 0; ks < 36; ++ks) {
        // wave-uniform tap position
        const int khw = ks >> 2;               // 0..8
        const int kh  = khw / 3;
        const int kw  = khw - kh * 3;
        // per-lane input channel pair
        const int ic  = ((ks & 3) << 2) + (kp << 1);   // even -> 8B aligned

        const int iy0 = 2 * y0 + kh, ix0 = 2 * x0 + kw;  // padded coords
        const int iy1 = 2 * y1 + kh, ix1 = 2 * x1 + kw;

        v2f a0 = *(const v2f*)(ap0 + ks * 64);
        v2f a1 = *(const v2f*)(ap1 + ks * 64);
        v2f b0 = *(const v2f*)(xb + ((iy0 * 130 + ix0) * 16 + ic));
        v2f b1 = *(const v2f*)(xb + ((iy1 * 130 + ix1) * 16 + ic));

        acc00 = __builtin_amdgcn_wmma_f32_16x16x4_f32(false, a0, false, b0,
                                                      (short)0, acc00, false, false);
        acc01 = __builtin_amdgcn_wmma_f32_16x16x4_f32(false, a0, false, b1,
                                                      (short)0, acc01, false, false);
        acc10 = __builtin_amdgcn_wmma_f32_16x16x4_f32(false, a1, false, b0,
                                                      (short)0, acc10, false, false);
        acc11 = __builtin_amdgcn_wmma_f32_16x16x4_f32(false, a1, false, b1,
                                                      (short)0, acc11, false, false);
    }

    // scale = exp(0.5*theta) / sqrt(N_rf),  N_rf = 32*64*64 = 131072
    const float scale = __expf(0.5f * theta[0]) * 0.00276213586400995f;

    const int chb = cg * 32 + (kp << 3);       // base channel for this lane half
#pragma unroll
    for (int v = 0; v < 8; ++v) {
        // tile (ct0, pos0)
        {
            int ch = chb + v;
            int mc = ch >> 5, o = ch & 31;
            size_t base = ((size_t)(b * 640 + mc * 64 + o) << 12) + (y0 << 6) + x0;
            float ph = acc00[v];
            out[base]               = scale * __cosf(ph);
            out[base + (32 << 12)]  = scale * __sinf(ph);
        }
        // tile (ct1, pos0)
        {
            int ch = chb + 16 + v;
            int mc = ch >> 5, o = ch & 31;
            size_t base = ((size_t)(b * 640 + mc * 64 + o) << 12) + (y0 << 6) + x0;
            float ph = acc10[v];
            out[base]               = scale * __cosf(ph);
            out[base + (32 << 12)]  = scale * __sinf(ph);
        }
        // tile (ct0, pos1)
        {
            int ch = chb + v;
            int mc = ch >> 5, o = ch & 31;
            size_t base = ((size_t)(b * 640 + mc * 64 + o) << 12) + (y1 << 6) + x1;
            float ph = acc01[v];
            out[base]               = scale * __cosf(ph);
            out[base + (32 << 12)]  = scale * __sinf(ph);
        }
        // tile (ct1, pos1)
        {
            int ch = chb + 16 + v;
            int mc = ch >> 5, o = ch & 31;
            size_t base = ((size_t)(b * 640 + mc * 64 + o) << 12) + (y1 << 6) + x1;
            float ph = acc11[v];
            out[base]               = scale * __cosf(ph);
            out[base + (32 << 12)]  = scale * __sinf(ph);
        }
    }
}

// ---------------------------------------------------------------------------
// Host launcher
// ---------------------------------------------------------------------------
extern "C" void kernel_launch(void* const* d_in, const int* in_sizes, int n_in,
                              void* d_out, int out_size, void* d_ws, size_t ws_size,
                              hipStream_t stream) {
    const float* x     = (const float*)d_in[0];  // (8,16,128,128)
    const float* theta = (const float*)d_in[1];  // (1,)
    const float* Wmean = (const float*)d_in[2];  // (144,32)
    const float* Wlogs = (const float*)d_in[3];  // (144,32)
    const float* Weps  = (const float*)d_in[4];  // (10,144,32)
    float* out = (float*)d_out;                  // (8,640,64,64)

    float* xpad  = (float*)d_ws;                 // 8*130*130*16 fp32
    float* apack = xpad + XPAD_ELEMS;            // 20*36*32 float2

    rff_pad_nhwc<<<(8 * 130 * 130 + 255) / 256, 256, 0, stream>>>(x, xpad);
    rff_pack_w<<<(APACK_F2 + 255) / 256, 256, 0, stream>>>(Wmean, Wlogs, Weps, apack);
    rff_conv_main<<<1280, 256, 0, stream>>>(xpad, apack, theta, out);
}